// NeuralSplineFourierFilter_5162550689963
// MI455X (gfx1250) — compile-verified
//
#include <hip/hip_runtime.h>
#include <math.h>

// ---------------------------------------------------------------------------
// NeuralSplineFourierFilter for MI455X (gfx1250)
//
// Phase 1 (1 wave): tiny MLP via V_WMMA_F32_16X16X4_F32, then serial build of
//   a per-interval De Boor table: for each of the 7 possible knot intervals
//   s = k-3, precompute the 6 alpha coefficients (inv, -t_lo*inv) and the 4
//   control points.  alpha = fma(xq, inv, c) -> 1 FMA per alpha.
// Phase 2 (bulk, HBM-bound): streaming float4 kernel, ~26 VALU + 4 conflict-
//   free ds_load_b128 per element; non-temporal b128 store.
// ---------------------------------------------------------------------------

typedef float v2f __attribute__((ext_vector_type(2)));
typedef float v4f __attribute__((ext_vector_type(4)));
typedef float v8f __attribute__((ext_vector_type(8)));

#define TBL_STRIDE 24   // floats per interval slot; 24 mod 64-bank pattern => no conflicts

__global__ __launch_bounds__(32) void nsff_setup(
    const float* __restrict__ a,
    const float* __restrict__ W1, const float* __restrict__ b1,
    const float* __restrict__ W2, const float* __restrict__ b2,
    const float* __restrict__ Ww, const float* __restrict__ bw,
    const float* __restrict__ Wk, const float* __restrict__ bk,
    float* __restrict__ params)
{
    __shared__ float cat[16];
    const int lane = threadIdx.x;       // 0..31, one full wave (EXEC all 1s at WMMA)
    const int col  = lane & 15;

    const float av = a[0];
    // h1 = sin(a*W1 + b1); valid on all 32 lanes (upper half duplicates cols)
    float h1 = __builtin_sinf(av * W1[col] + b1[col]);

    // ---- h2pre = h1 @ W2 via 4x V_WMMA_F32_16X16X4_F32 --------------------
    // A (16x4 f32): lanes 0-15 rows M; VGPR0 = K even, VGPR1 = K odd; upper
    // half-wave carries K+2.  We broadcast h1 into every row of A, so every
    // row of D equals h1 @ W2 and acc[0] is the answer on all lanes.
    v8f acc = {0.f,0.f,0.f,0.f,0.f,0.f,0.f,0.f};
#pragma unroll
    for (int q = 0; q < 4; ++q) {
        const int rA = 4*q + ((lane < 16) ? 0 : 2);
        v2f A, B;
        A.x = __shfl(h1, 4*q + ((lane < 16) ? 0 : 2), 32);
        A.y = __shfl(h1, 4*q + ((lane < 16) ? 1 : 3), 32);
        B.x = W2[rA * 16 + col];        // B (4x16): row K striped across lanes
        B.y = W2[(rA + 1) * 16 + col];
        acc = __builtin_amdgcn_wmma_f32_16x16x4_f32(false, A, false, B,
                                                    (short)0, acc, false, false);
    }
    float h2 = __builtin_sinf(acc[0] + b2[col]);

    // ---- fused heads: B = [Ww (16x9) | Wk (16x7)] -> one more 16x16x16 ----
    v8f acc2 = {0.f,0.f,0.f,0.f,0.f,0.f,0.f,0.f};
#pragma unroll
    for (int q = 0; q < 4; ++q) {
        const int rA = 4*q + ((lane < 16) ? 0 : 2);
        v2f A, B;
        A.x = __shfl(h2, 4*q + ((lane < 16) ? 0 : 2), 32);
        A.y = __shfl(h2, 4*q + ((lane < 16) ? 1 : 3), 32);
        B.x = (col < 9) ? Ww[rA * 9 + col]       : Wk[rA * 7 + (col - 9)];
        B.y = (col < 9) ? Ww[(rA + 1) * 9 + col] : Wk[(rA + 1) * 7 + (col - 9)];
        acc2 = __builtin_amdgcn_wmma_f32_16x16x4_f32(false, A, false, B,
                                                     (short)0, acc2, false, false);
    }
    float ov = acc2[0] + ((col < 9) ? bw[col] : bk[col - 9]);
    if (lane < 16) cat[col] = ov;
    __syncthreads();

    // ---- serial epilogue on lane 0: softmax, cumsum, knot/coef tables -----
    if (lane == 0) {
        float w[10];
        w[0] = 0.f;
        for (int i = 0; i < 9; ++i) w[i + 1] = cat[i];
        float kr[7];
        for (int i = 0; i < 7; ++i) kr[i] = cat[9 + i];

        float m = kr[0];
        for (int i = 1; i < 7; ++i) m = fmaxf(m, kr[i]);
        float e[7], sum = 0.f;
        for (int i = 0; i < 7; ++i) { e[i] = __builtin_expf(kr[i] - m); sum += e[i]; }
        const float inv_sum = 1.f / sum;

        float t[14];
        t[0] = t[1] = t[2] = t[3] = 0.f;
        float run = 0.f;
        for (int i = 0; i < 7; ++i) { run += e[i] * inv_sum; t[4 + i] = run; }
        t[11] = t[12] = t[13] = 1.f;

        // De Boor (p=3) step -> (lo,hi) knot offsets relative to s = k-3:
        // r=1: (3,6)(2,5)(1,4); r=2: (3,5)(2,4); r=3: (3,4)
        const int LO[6] = {3, 2, 1, 3, 2, 3};
        const int HI[6] = {6, 5, 4, 5, 4, 4};
        for (int s = 0; s < 7; ++s) {
            for (int p = 0; p < 6; ++p) {
                const float lo = t[s + LO[p]], hi = t[s + HI[p]];
                const float inv = 1.f / (hi - lo);
                params[s * 16 + 2 * p]     = inv;
                params[s * 16 + 2 * p + 1] = -lo * inv;
            }
            for (int j = 0; j < 4; ++j) params[s * 16 + 12 + j] = w[s + j];
        }
        // interval thresholds c1..c6 = interior knots kk[1..6]
        for (int i = 0; i < 6; ++i) params[112 + i] = t[4 + i];
    }
}

__global__ __launch_bounds__(256) void nsff_eval(
    const float* __restrict__ x, float* __restrict__ out,
    const float* __restrict__ params, long long n)
{
    __shared__ float tbl[7 * TBL_STRIDE];
    const int tid = threadIdx.x;
    if (tid < 112) tbl[(tid >> 4) * TBL_STRIDE + (tid & 15)] = params[tid];
    // uniform addresses -> scalar loads, broadcast to wave
    const float c1 = params[112], c2 = params[113], c3 = params[114];
    const float c4 = params[115], c5 = params[116], c6 = params[117];
    __syncthreads();

    const long long base =
        ((long long)blockIdx.x * blockDim.x + threadIdx.x) * 4;
    if (base >= n) return;

    float xv[4];
    const bool full = (base + 3 < n);
    if (full) {
        const v4f v = *(const v4f*)(x + base);
        xv[0] = v.x; xv[1] = v.y; xv[2] = v.z; xv[3] = v.w;
    } else {
        for (int j = 0; j < 4; ++j) xv[j] = (base + j < n) ? x[base + j] : 0.f;
    }

    float r[4];
#pragma unroll
    for (int j = 0; j < 4; ++j) {
        const float xq =
            fminf(fmaxf(xv[j] * 0.57735026918962576f, 0.0f), 0.9999f);
        // interval index: s = #{interior knots <= xq}, s in [0,6]
        const int s = (xq >= c1) + (xq >= c2) + (xq >= c3) +
                      (xq >= c4) + (xq >= c5) + (xq >= c6);
        const float* T = &tbl[s * TBL_STRIDE];
        const v4f q0 = *(const v4f*)(T);
        const v4f q1 = *(const v4f*)(T + 4);
        const v4f q2 = *(const v4f*)(T + 8);
        const v4f q3 = *(const v4f*)(T + 12);
        // one FMA per alpha (denominators pre-inverted, -t_lo*inv folded in)
        const float a33 = fmaf(xq, q0.x, q0.y);
        const float a23 = fmaf(xq, q0.z, q0.w);
        const float a13 = fmaf(xq, q1.x, q1.y);
        const float a32 = fmaf(xq, q1.z, q1.w);
        const float a22 = fmaf(xq, q2.x, q2.y);
        const float a31 = fmaf(xq, q2.z, q2.w);
        float d0 = q3.x, d1 = q3.y, d2 = q3.z, d3 = q3.w;
        d3 = fmaf(a33, d3 - d2, d2);   // r=1, j=3
        d2 = fmaf(a23, d2 - d1, d1);   // r=1, j=2
        d1 = fmaf(a13, d1 - d0, d0);   // r=1, j=1
        d3 = fmaf(a32, d3 - d2, d2);   // r=2, j=3
        d2 = fmaf(a22, d2 - d1, d1);   // r=2, j=2
        d3 = fmaf(a31, d3 - d2, d2);   // r=3, j=3
        r[j] = d3;
    }

    if (full) {
        v4f o; o.x = r[0]; o.y = r[1]; o.z = r[2]; o.w = r[3];
        __builtin_nontemporal_store(o, (v4f*)(out + base));  // single-use stream
    } else {
        for (int j = 0; j < 4; ++j)
            if (base + j < n) out[base + j] = r[j];
    }
}

extern "C" void kernel_launch(void* const* d_in, const int* in_sizes, int n_in,
                              void* d_out, int out_size, void* d_ws, size_t ws_size,
                              hipStream_t stream) {
    const float* x  = (const float*)d_in[0];
    const float* a  = (const float*)d_in[1];
    const float* W1 = (const float*)d_in[2];
    const float* b1 = (const float*)d_in[3];
    const float* W2 = (const float*)d_in[4];
    const float* b2 = (const float*)d_in[5];
    const float* Ww = (const float*)d_in[6];
    const float* bw = (const float*)d_in[7];
    const float* Wk = (const float*)d_in[8];
    const float* bk = (const float*)d_in[9];
    float* params = (float*)d_ws;        // 118 floats used
    float* out    = (float*)d_out;

    nsff_setup<<<1, 32, 0, stream>>>(a, W1, b1, W2, b2, Ww, bw, Wk, bk, params);

    const long long n  = (long long)in_sizes[0];
    const long long n4 = (n + 3) >> 2;
    const int block = 256;
    const long long grid = (n4 + block - 1) / block;
    nsff_eval<<<(unsigned)grid, block, 0, stream>>>(x, out, params, n);
}